// CorrectAndSmooth_8735963480515
// MI455X (gfx1250) — compile-verified
//
#include <hip/hip_runtime.h>
#include <stdint.h>

// Problem constants (fixed by the reference)
#define NN   100000     // nodes
#define NE   1600000    // edges
#define NC   40         // classes
#define NTR  50000      // train / mask size
#define C4   (NC / 4)   // float4 groups per row = 10

static constexpr float A_CORR   = 0.979f;
static constexpr float A_SMOOTH = 0.756f;

#define NPB   32                 // nodes per block in aggregate
#define AGGT  (NPB * C4)         // 320 threads = 10 wave32
#define CHUNK 2048               // ints staged in LDS per iteration (8 KB)
#define TPB   256

typedef __attribute__((ext_vector_type(2))) float v2f;
typedef __attribute__((ext_vector_type(8))) float v8f;

// ---------------- setup kernels ----------------

__global__ void k_zero_i32(int* p, int n) {
  int i = blockIdx.x * blockDim.x + threadIdx.x;
  if (i < n) p[i] = 0;
}
__global__ void k_zero_f32(float* p, int n) {
  int i = blockIdx.x * blockDim.x + threadIdx.x;
  if (i < n) p[i] = 0.f;
}
__global__ void k_count(const int* __restrict__ dst, int* __restrict__ deg) {
  int e = blockIdx.x * blockDim.x + threadIdx.x;
  if (e < NE) atomicAdd(&deg[dst[e]], 1);
}
// per-block exclusive scan (1024 elems/block) + block totals
__global__ void k_scan_block(const int* __restrict__ in, int* __restrict__ excl,
                             int* __restrict__ bsum, int n) {
  __shared__ int s[1024];
  int t = threadIdx.x;
  int i = blockIdx.x * 1024 + t;
  int x = (i < n) ? in[i] : 0;
  s[t] = x;
  __syncthreads();
  for (int off = 1; off < 1024; off <<= 1) {
    int v = (t >= off) ? s[t - off] : 0;
    __syncthreads();
    s[t] += v;
    __syncthreads();
  }
  if (i < n) excl[i] = s[t] - x;
  if (t == 1023) bsum[blockIdx.x] = s[1023];
}
__global__ void k_scan_sums(int* bsum, int nb, int* row_ptr_last) {
  if (blockIdx.x == 0 && threadIdx.x == 0) {
    int run = 0;
    for (int b = 0; b < nb; ++b) { int v = bsum[b]; bsum[b] = run; run += v; }
    *row_ptr_last = run;   // row_ptr[NN] = total edges
  }
}
__global__ void k_add_offsets(const int* __restrict__ excl, const int* __restrict__ bsum,
                              int* __restrict__ row_ptr, int* __restrict__ cursor, int n) {
  int i = blockIdx.x * blockDim.x + threadIdx.x;
  if (i < n) {
    int v = excl[i] + bsum[i >> 10];
    row_ptr[i] = v;
    cursor[i]  = v;
  }
}
__global__ void k_fill_csr(const int* __restrict__ src, const int* __restrict__ dst,
                           int* __restrict__ cursor, int* __restrict__ col) {
  int e = blockIdx.x * blockDim.x + threadIdx.x;
  if (e < NE) {
    int pos = atomicAdd(&cursor[dst[e]], 1);
    col[pos] = src[e];
  }
}
__global__ void k_norm(const int* __restrict__ deg, float* __restrict__ nrm) {
  int v = blockIdx.x * blockDim.x + threadIdx.x;
  if (v < NN) nrm[v] = rsqrtf(fmaxf((float)deg[v], 1.f));
}

// ---------------- phase kernels ----------------

__global__ void k_init_error(const float* __restrict__ y_soft, const int* __restrict__ y_true,
                             const int* __restrict__ mask, float* __restrict__ err) {
  int t = blockIdx.x * blockDim.x + threadIdx.x;
  if (t >= NTR * NC) return;
  int i = t / NC, c = t - i * NC;
  int m = mask[i];
  float oh = (c == y_true[i]) ? 1.f : 0.f;
  err[m * NC + c] = oh - y_soft[m * NC + c];
}
__global__ void k_sigma(const float* __restrict__ err, const int* __restrict__ mask,
                        float* __restrict__ sigma) {
  __shared__ float s[TPB];
  int t = blockIdx.x * TPB + threadIdx.x;
  float v = 0.f;
  if (t < NTR * NC) {
    int i = t / NC, c = t - i * NC;
    v = fabsf(err[mask[i] * NC + c]);
  }
  s[threadIdx.x] = v;
  __syncthreads();
  for (int off = TPB / 2; off > 0; off >>= 1) {
    if (threadIdx.x < off) s[threadIdx.x] += s[threadIdx.x + off];
    __syncthreads();
  }
  if (threadIdx.x == 0) atomicAdd(sigma, s[0]);
}
__global__ void k_scale_copy(const float* __restrict__ y, float* __restrict__ lastb,
                             float coef, int n) {
  int i = blockIdx.x * blockDim.x + threadIdx.x;
  if (i < n) lastb[i] = coef * y[i];
}
// h[v][c] = y[v][c] * norm[v]   (pre-scaled propagation state)
__global__ void k_prescale(const float* __restrict__ y, const float* __restrict__ nrm,
                           float* __restrict__ h) {
  int t = blockIdx.x * blockDim.x + threadIdx.x;
  if (t < NN * NC) h[t] = y[t] * nrm[t / NC];
}

// Hot kernel: one propagation layer in pre-scaled h-space.
//   acc[v][c] = sum_{u in col[row_ptr[v]:row_ptr[v+1]]} h_in[u][c]
//   y_new     = clamp(last + alpha * acc * norm[v], clo, chi)
//   h_out     = y_new * norm[v]           (if h_out != null)
//   y_out     = y_new                     (if y_out != null)
// CSR slice for this block's 32 nodes is contiguous; staged into LDS with the
// CDNA5 async data mover (global_load_async_to_lds_b128 + s_wait_asynccnt).
__global__ __launch_bounds__(AGGT)
void k_aggregate(const float4* __restrict__ h_in, const float* __restrict__ nrm,
                 const float4* __restrict__ lastb, float4* __restrict__ y_out,
                 float4* __restrict__ h_out,
                 const int* __restrict__ row_ptr, const int* __restrict__ col,
                 float alpha, float clo, float chi) {
  __shared__ __align__(16) int s_col[CHUNK];
  __shared__ int s_rp[NPB + 1];
  const int tid  = threadIdx.x;
  const int vloc = tid / C4;
  const int q    = tid - vloc * C4;
  const int vbase = blockIdx.x * NPB;
  if (tid <= NPB) s_rp[tid] = row_ptr[min(vbase + tid, NN)];
  __syncthreads();
  const int base = s_rp[0];
  const int end  = s_rp[NPB];
  const int v    = vbase + vloc;
  const int my_s = s_rp[vloc];
  const int my_e = s_rp[vloc + 1];
  float4 acc = make_float4(0.f, 0.f, 0.f, 0.f);

  const int ch0  = base & ~3;          // 16B-aligned global start
  const int endA = (end + 3) & ~3;     // col[] is padded past NE
  for (int chunk = ch0; chunk < endA; chunk += CHUNK) {   // block-uniform loop
    const int cntA = min(CHUNK, endA - chunk);
    const int n4 = cntA >> 2;
    for (int i = tid; i < n4; i += AGGT) {
      uint32_t lds  = (uint32_t)(uintptr_t)(&s_col[i << 2]);        // LDS byte addr (low 32b)
      uint32_t goff = (uint32_t)((chunk + (i << 2)) * 4);           // byte offset from col base
      asm volatile("global_load_async_to_lds_b128 %0, %1, %2"
                   :: "v"(lds), "v"(goff), "s"(col) : "memory");
    }
    asm volatile("s_wait_asynccnt 0" ::: "memory");   // this wave's LDS writes done
    __syncthreads();                                  // all waves' writes visible
    const int lo = max(my_s, chunk);
    const int hi = min(my_e, chunk + cntA);
#pragma unroll 4
    for (int e = lo; e < hi; ++e) {
      int u = s_col[e - chunk];
      float4 f = h_in[u * C4 + q];
      acc.x += f.x;
      acc.y += f.y;
      acc.z += f.z;
      acc.w += f.w;
    }
    __syncthreads();                                  // before s_col is reused
  }
  if (v < NN) {
    float nv = nrm[v];
    float an = alpha * nv;
    float4 l = lastb[v * C4 + q];
    float4 r;
    r.x = fminf(fmaxf(fmaf(acc.x, an, l.x), clo), chi);
    r.y = fminf(fmaxf(fmaf(acc.y, an, l.y), clo), chi);
    r.z = fminf(fmaxf(fmaf(acc.z, an, l.z), clo), chi);
    r.w = fminf(fmaxf(fmaf(acc.w, an, l.w), clo), chi);
    if (y_out) y_out[v * C4 + q] = r;
    if (h_out) {
      float4 hh;
      hh.x = r.x * nv; hh.y = r.y * nv; hh.z = r.z * nv; hh.w = r.w * nv;
      h_out[v * C4 + q] = hh;
    }
  }
}

// Autoscale + apply: denom[v] = sum_c |sm[v][c]| is a GEMV with a ones vector
// (denom = |SM| . 1), the one matrix-shaped op in the pipeline -> matrix pipe.
// One wave32 per 16-node tile; A = 16x4 |sm| chunk, B = 4x16 ones,
// 10 chained V_WMMA_F32_16X16X4_F32 accumulate row sums into every D column.
__global__ __launch_bounds__(32)
void k_apply_correction(const float* __restrict__ y_soft, const float* __restrict__ sm,
                        const float* __restrict__ sigma_sum, float* __restrict__ y_out) {
  const int lane = threadIdx.x;          // 0..31, full wave (EXEC all ones)
  const int v0 = blockIdx.x * 16;        // NN % 16 == 0
  __shared__ float s_denom[16];
#if defined(__gfx1250__) && __has_builtin(__builtin_amdgcn_wmma_f32_16x16x4_f32)
  {
    const int m = lane & 15;             // A row (ISA A layout: lanes 0-15 / 16-31)
    const int khalf = lane >> 4;         // 0 -> K{0,1}, 1 -> K{2,3}
    const float* row = sm + (size_t)(v0 + m) * NC + 2 * khalf;
    v8f d = {0.f, 0.f, 0.f, 0.f, 0.f, 0.f, 0.f, 0.f};
    v2f b; b.x = 1.f; b.y = 1.f;         // ones B-matrix (4x16)
#pragma unroll
    for (int kb = 0; kb < NC; kb += 4) {
      v2f a;
      a.x = fabsf(row[kb]);              // K = kb + 2*khalf
      a.y = fabsf(row[kb + 1]);          // K = kb + 2*khalf + 1
      d = __builtin_amdgcn_wmma_f32_16x16x4_f32(false, a, false, b,
                                                (short)0, d, false, false);
    }
    // D layout: VGPR r holds M=r (lanes 0-15, N=lane) / M=r+8 (lanes 16-31).
    // Column 0 lives in lanes 0 and 16 -> they spill the 16 row sums.
    if ((lane & 15) == 0) {
      const int mb = khalf * 8;
#pragma unroll
      for (int r = 0; r < 8; ++r) s_denom[mb + r] = d[r];
    }
  }
#else
  if (lane < 16) {
    const float* row = sm + (size_t)(v0 + lane) * NC;
    float t = 0.f;
    for (int c = 0; c < NC; ++c) t += fabsf(row[c]);
    s_denom[lane] = t;
  }
#endif
  __syncthreads();
  const float sig = (*sigma_sum) / (float)NTR;
  const size_t base = (size_t)v0 * NC;
  for (int j = lane; j < 16 * NC; j += 32) {   // coalesced tile update
    float scale = sig / s_denom[j / NC];
    if (isinf(scale) || scale > 1000.f) scale = 1.f;
    y_out[base + j] = fmaf(scale, sm[base + j], y_soft[base + j]);
  }
}

__global__ void k_set_mask(float* __restrict__ y, const int* __restrict__ mask,
                           const int* __restrict__ y_true) {
  int t = blockIdx.x * blockDim.x + threadIdx.x;
  if (t >= NTR * NC) return;
  int i = t / NC, c = t - i * NC;
  y[mask[i] * NC + c] = (c == y_true[i]) ? 1.f : 0.f;
}

// ---------------- host orchestration ----------------

static inline size_t align256(size_t x) { return (x + 255) & ~(size_t)255; }
#define GRID(n) (((n) + TPB - 1) / TPB)

extern "C" void kernel_launch(void* const* d_in, const int* in_sizes, int n_in,
                              void* d_out, int out_size, void* d_ws, size_t ws_size,
                              hipStream_t stream) {
  (void)in_sizes; (void)n_in; (void)out_size; (void)ws_size;
  const float* y_soft = (const float*)d_in[0];
  const int*   src    = (const int*)d_in[1];
  const int*   dst    = (const int*)d_in[2];
  const int*   y_true = (const int*)d_in[3];
  const int*   mask   = (const int*)d_in[4];
  float*       out    = (float*)d_out;

  char* ws = (char*)d_ws;
  size_t off = 0;
  auto take = [&](size_t bytes) { void* p = ws + off; off = align256(off + bytes); return p; };
  int*   deg    = (int*)  take((size_t)NN * 4);
  int*   excl   = (int*)  take((size_t)NN * 4);
  int*   bsum   = (int*)  take(128 * 4);
  int*   rowp   = (int*)  take((size_t)(NN + 1) * 4);
  int*   cursor = (int*)  take((size_t)NN * 4);
  int*   col    = (int*)  take((size_t)(NE + 16) * 4);   // padded for b128 round-up
  float* nrm    = (float*)take((size_t)NN * 4);
  float* sigma  = (float*)take(256);
  float* lastb  = (float*)take((size_t)NN * NC * 4);
  float* bufA   = (float*)take((size_t)NN * NC * 4);
  float* bufB   = (float*)take((size_t)NN * NC * 4);

  const int NB_SCAN = (NN + 1023) / 1024;   // 98

  // ---- CSR by dst + norm (rebuilt every call: deterministic) ----
  k_zero_i32<<<GRID(NN), TPB, 0, stream>>>(deg, NN);
  k_count<<<GRID(NE), TPB, 0, stream>>>(dst, deg);
  k_scan_block<<<NB_SCAN, 1024, 0, stream>>>(deg, excl, bsum, NN);
  k_scan_sums<<<1, 1, 0, stream>>>(bsum, NB_SCAN, rowp + NN);
  k_add_offsets<<<GRID(NN), TPB, 0, stream>>>(excl, bsum, rowp, cursor, NN);
  k_fill_csr<<<GRID(NE), TPB, 0, stream>>>(src, dst, cursor, col);
  k_norm<<<GRID(NN), TPB, 0, stream>>>(deg, nrm);

  const int AGG_GRID = (NN + NPB - 1) / NPB;   // 3125

  // ---- correct phase ----
  // bufA = error (y-space); sigma + last captured up front, then bufA is recycled.
  k_zero_f32<<<GRID(NN * NC), TPB, 0, stream>>>(bufA, NN * NC);
  k_init_error<<<GRID(NTR * NC), TPB, 0, stream>>>(y_soft, y_true, mask, bufA);
  k_zero_f32<<<1, 1, 0, stream>>>(sigma, 1);
  k_sigma<<<GRID(NTR * NC), TPB, 0, stream>>>(bufA, mask, sigma);
  k_scale_copy<<<GRID(NN * NC), TPB, 0, stream>>>(bufA, lastb, 1.f - A_CORR, NN * NC);
  k_prescale<<<GRID(NN * NC), TPB, 0, stream>>>(bufA, nrm, bufB);   // h0 = error*norm
  float* hcur = bufB; float* hnxt = bufA;                           // bufA now free
  for (int i = 0; i < 10; ++i) {
    const bool lastL = (i == 9);
    // layers 0..8 carry h only; layer 9 materializes y (= smoothed_error)
    k_aggregate<<<AGG_GRID, AGGT, 0, stream>>>(
        (const float4*)hcur, nrm, (const float4*)lastb,
        lastL ? (float4*)hnxt : (float4*)nullptr,   // y_out
        lastL ? (float4*)nullptr : (float4*)hnxt,   // h_out
        rowp, col, A_CORR, -1.f, 1.f);
    float* t = hcur; hcur = hnxt; hnxt = t;
  }
  float* sm    = hcur;   // smoothed_error (y-space)
  float* ybuf  = hnxt;   // free buffer
  k_apply_correction<<<NN / 16, 32, 0, stream>>>(y_soft, sm, sigma, ybuf);
  k_set_mask<<<GRID(NTR * NC), TPB, 0, stream>>>(ybuf, mask, y_true);
  k_scale_copy<<<GRID(NN * NC), TPB, 0, stream>>>(ybuf, lastb, 1.f - A_SMOOTH, NN * NC);
  k_prescale<<<GRID(NN * NC), TPB, 0, stream>>>(ybuf, nrm, sm);     // h0 = y*norm (sm dead)

  // ---- smooth phase (final layer writes y straight into d_out) ----
  hcur = sm; hnxt = ybuf;
  for (int i = 0; i < 10; ++i) {
    const bool lastL = (i == 9);
    k_aggregate<<<AGG_GRID, AGGT, 0, stream>>>(
        (const float4*)hcur, nrm, (const float4*)lastb,
        lastL ? (float4*)out : (float4*)nullptr,    // y_out
        lastL ? (float4*)nullptr : (float4*)hnxt,   // h_out
        rowp, col, A_SMOOTH, 0.f, 1.f);
    float* t = hcur; hcur = hnxt; hnxt = t;
  }
}